// GNNPolicy_multimodal_58884001628294
// MI455X (gfx1250) — compile-verified
//
#include <hip/hip_runtime.h>
#include <hip/hip_bf16.h>

// ---------------------------------------------------------------------------
// GNN policy forward for MI455X (gfx1250, wave32).
//
// Math restructuring vs reference (exact, by linearity):
//  * LN over the single edge feature == edge_ln_b[0] exactly -> constant vec.
//  * Per-edge linears hoisted to per-node GEMMs + gathers.
//  * W_final commutes with segment_sum: scatter-add ReLU(LN(h)) and degree,
//    then one GEMM  F = agg @ Wf^T + deg (x) bf.
// Dense GEMMs ([N,64] x [64,K]^T) use V_WMMA_F32_16X16X4_F32.
// ---------------------------------------------------------------------------

typedef __attribute__((ext_vector_type(2))) float v2f;
typedef __attribute__((ext_vector_type(8))) float v8f;

__device__ __forceinline__ void atomAddF(float* p, float v) {
  // lowers to global_atomic_add_f32 (no return) on gfx1250
  unsafeAtomicAdd(p, v);
}

// ------------------------------- utility -----------------------------------
__global__ void k_fill(float* __restrict__ p, long long n, float v) {
  long long i = (long long)blockIdx.x * blockDim.x + threadIdx.x;
  if (i < n) p[i] = v;
}

// evec[j] = edge_ln_b[0] * W_edge[j][0]   (exact value of W_edge @ LN(ef))
__global__ void k_evec(const float* __restrict__ elnb,
                       const float* __restrict__ edgeW,
                       float* __restrict__ evec) {
  int j = threadIdx.x;
  if (j < 64) evec[j] = elnb[0] * edgeW[j];
}

// ------------------------- input embedding stage 1 --------------------------
// y[n][ch] = ReLU( b1[ch] + sum_i W1[ch][i] * LN(x[n])[i] ),  nf in {4,6}
__global__ void k_embed1(const float* __restrict__ x, int nf, int N,
                         const float* __restrict__ g, const float* __restrict__ b,
                         const float* __restrict__ W1, const float* __restrict__ b1,
                         float* __restrict__ y) {
  int u = blockIdx.x * blockDim.x + threadIdx.x;
  int node = u >> 6, ch = u & 63;
  if (node >= N) return;
  float xs[8];
  float mu = 0.f;
  for (int i = 0; i < nf; ++i) { xs[i] = x[(size_t)node * nf + i]; mu += xs[i]; }
  mu /= (float)nf;
  float var = 0.f;
  for (int i = 0; i < nf; ++i) { float d = xs[i] - mu; var += d * d; }
  var /= (float)nf;
  float inv = rsqrtf(var + 1e-5f);
  float acc = b1[ch];
  for (int i = 0; i < nf; ++i)
    acc += W1[(size_t)ch * nf + i] * ((xs[i] - mu) * inv * g[i] + b[i]);
  y[(size_t)node * 64 + ch] = fmaxf(acc, 0.f);
}

// ------------------------------ WMMA GEMM ----------------------------------
// Y[N,64] = act( X[N,K] @ W[64,K]^T + bias (+ deg[row]*bias2[col]) )
// One wave -> 16 rows x 64 cols, K stepped by 4 via v_wmma_f32_16x16x4_f32.
// A frag (16x4): lane<16 holds M=lane K={k,k+1}; lane>=16 holds K={k+2,k+3}.
// B frag (4x16): vgpr0 = rows K=k (lanes 0-15) / K=k+2 (lanes 16-31), N=lane&15.
// C/D (16x16):   vgpr i -> row i (lanes 0-15) / row i+8 (lanes 16-31), col=lane&15.
__global__ void __launch_bounds__(256)
k_gemm64(const float* __restrict__ X, int N, int K,
         const float* __restrict__ W, const float* __restrict__ bias,
         const float* __restrict__ deg, const float* __restrict__ bias2,
         float* __restrict__ Y, int relu) {
  const int wave = threadIdx.x >> 5;
  const int lane = threadIdx.x & 31;
  const int l15  = lane & 15;
  const int hi   = lane >> 4;           // 0 or 1
  const int row0 = (blockIdx.x * 8 + wave) * 16;
  if (row0 >= N) return;
  const int rowA   = row0 + l15;
  const bool rowok = rowA < N;
  const float* xrow = X + (size_t)(rowok ? rowA : 0) * K;

  v8f acc[4] = {};
  for (int k = 0; k < K; k += 4) {
    v2f a;
    {
      const float2 av = *(const float2*)(xrow + k + 2 * hi);
      a.x = rowok ? av.x : 0.f;
      a.y = rowok ? av.y : 0.f;
    }
#pragma unroll
    for (int nt = 0; nt < 4; ++nt) {
      const float2 wv = *(const float2*)(W + (size_t)(nt * 16 + l15) * K + k + 2 * hi);
      v2f bf; bf.x = wv.x; bf.y = wv.y;   // B[kk][n] = W[n][kk]
      acc[nt] = __builtin_amdgcn_wmma_f32_16x16x4_f32(
          false, a, false, bf, (short)0, acc[nt], false, false);
    }
  }
#pragma unroll
  for (int nt = 0; nt < 4; ++nt) {
    const int col = nt * 16 + l15;
    const float bv  = bias  ? bias[col]  : 0.f;
    const float b2v = bias2 ? bias2[col] : 0.f;
#pragma unroll
    for (int i = 0; i < 8; ++i) {
      const int row = row0 + i + 8 * hi;
      if (row < N) {
        float val = acc[nt][i] + bv;
        if (deg) val += deg[row] * b2v;
        if (relu) val = fmaxf(val, 0.f);
        Y[(size_t)row * 64 + col] = val;
      }
    }
  }
}

// ------------------------------ edge kernel --------------------------------
// 8 lanes per edge; per edge:  h = L[tgt] + R[oth] + evec ; g=ReLU(LN(h)) ;
// agg[tgt]+=g (f32 atomics, L2-resident) ; deg[tgt]+=1.
__global__ void __launch_bounds__(256)
k_edge(const int* __restrict__ srcI, const int* __restrict__ dstI, int NE,
       const float* __restrict__ Ltgt, const float* __restrict__ Rsrc,
       const float* __restrict__ evec,
       const float* __restrict__ g, const float* __restrict__ b,
       float* __restrict__ agg, float* __restrict__ deg, int tgt_is_src) {
  int u = blockIdx.x * blockDim.x + threadIdx.x;
  int e = u >> 3;
  if (e >= NE) return;
  int gi = threadIdx.x & 7;
  int si = srcI[e], di = dstI[e];
  int tgt = tgt_is_src ? si : di;
  int oth = tgt_is_src ? di : si;
  const float4* lp = (const float4*)(Ltgt + (size_t)tgt * 64 + gi * 8);
  const float4* rp = (const float4*)(Rsrc + (size_t)oth * 64 + gi * 8);
  const float4* ep = (const float4*)(evec + gi * 8);
  float4 l0 = lp[0], l1 = lp[1], r0 = rp[0], r1 = rp[1], e0 = ep[0], e1 = ep[1];
  float h[8];
  h[0] = l0.x + r0.x + e0.x; h[1] = l0.y + r0.y + e0.y;
  h[2] = l0.z + r0.z + e0.z; h[3] = l0.w + r0.w + e0.w;
  h[4] = l1.x + r1.x + e1.x; h[5] = l1.y + r1.y + e1.y;
  h[6] = l1.z + r1.z + e1.z; h[7] = l1.w + r1.w + e1.w;
  float s = 0.f;
#pragma unroll
  for (int i = 0; i < 8; ++i) s += h[i];
  s += __shfl_xor(s, 1, 32); s += __shfl_xor(s, 2, 32); s += __shfl_xor(s, 4, 32);
  float mu = s * (1.f / 64.f);
  float vv = 0.f;
#pragma unroll
  for (int i = 0; i < 8; ++i) { float d = h[i] - mu; vv += d * d; }
  vv += __shfl_xor(vv, 1, 32); vv += __shfl_xor(vv, 2, 32); vv += __shfl_xor(vv, 4, 32);
  float inv = rsqrtf(vv * (1.f / 64.f) + 1e-5f);
  float* ap = agg + (size_t)tgt * 64 + gi * 8;
#pragma unroll
  for (int i = 0; i < 8; ++i) {
    int c = gi * 8 + i;
    float val = fmaxf((h[i] - mu) * inv * g[c] + b[c], 0.f);
    atomAddF(ap + i, val);
  }
  if (gi == 0) atomAddF(deg + tgt, 1.f);
}

// ------------------- LN(agg) || old  ->  cat[N,128] ------------------------
__global__ void __launch_bounds__(256)
k_lncat(const float* __restrict__ F, const float* __restrict__ oldf,
        const float* __restrict__ g, const float* __restrict__ b,
        float* __restrict__ cat, int N) {
  int u = blockIdx.x * blockDim.x + threadIdx.x;
  int row = u >> 3;
  if (row >= N) return;
  int gi = threadIdx.x & 7;
  const float4* fp = (const float4*)(F + (size_t)row * 64 + gi * 8);
  float4 f0 = fp[0], f1 = fp[1];
  float h[8] = {f0.x, f0.y, f0.z, f0.w, f1.x, f1.y, f1.z, f1.w};
  float s = 0.f;
#pragma unroll
  for (int i = 0; i < 8; ++i) s += h[i];
  s += __shfl_xor(s, 1, 32); s += __shfl_xor(s, 2, 32); s += __shfl_xor(s, 4, 32);
  float mu = s * (1.f / 64.f);
  float vv = 0.f;
#pragma unroll
  for (int i = 0; i < 8; ++i) { float d = h[i] - mu; vv += d * d; }
  vv += __shfl_xor(vv, 1, 32); vv += __shfl_xor(vv, 2, 32); vv += __shfl_xor(vv, 4, 32);
  float inv = rsqrtf(vv * (1.f / 64.f) + 1e-5f);
  float* cp = cat + (size_t)row * 128;
#pragma unroll
  for (int i = 0; i < 8; ++i) {
    int c = gi * 8 + i;
    cp[c] = (h[i] - mu) * inv * g[c] + b[c];
  }
  const float4* op = (const float4*)(oldf + (size_t)row * 64 + gi * 8);
  float4 o0 = op[0], o1 = op[1];
  ((float4*)(cp + 64 + gi * 8))[0] = o0;
  ((float4*)(cp + 64 + gi * 8))[1] = o1;
}

// ------------------------------ attention ----------------------------------
// H=4 heads, D=16, NT=128 keys. K/V staged in LDS (64KB of 320KB WGP LDS).
// One thread per (variable, head), online softmax (flash style).
__global__ void __launch_bounds__(256)
k_attn(const float* __restrict__ Q, const float* __restrict__ K,
       const float* __restrict__ V, float* __restrict__ O, int NVn) {
  __shared__ float sk[128 * 64];
  __shared__ float sv[128 * 64];
  for (int i = threadIdx.x; i < 128 * 64 / 4; i += 256) {
    ((float4*)sk)[i] = ((const float4*)K)[i];
    ((float4*)sv)[i] = ((const float4*)V)[i];
  }
  __syncthreads();
  int u = blockIdx.x * 256 + threadIdx.x;
  int var = u >> 2, head = u & 3;
  if (var >= NVn) return;
  float q[16];
  const float* qr = Q + (size_t)var * 64 + head * 16;
#pragma unroll
  for (int d = 0; d < 16; ++d) q[d] = qr[d];
  float m = -1e30f, l = 0.f, o[16] = {};
  for (int j = 0; j < 128; ++j) {
    const float* kr = sk + j * 64 + head * 16;
    float s = 0.f;
#pragma unroll
    for (int d = 0; d < 16; ++d) s += q[d] * kr[d];
    s *= 0.25f;  // 1/sqrt(16)
    float mn = fmaxf(m, s);
    float corr = __expf(m - mn);
    float p = __expf(s - mn);
    l = l * corr + p;
    const float* vr = sv + j * 64 + head * 16;
#pragma unroll
    for (int d = 0; d < 16; ++d) o[d] = o[d] * corr + p * vr[d];
    m = mn;
  }
  float invl = 1.f / l;
  float* orow = O + (size_t)var * 64 + head * 16;
#pragma unroll
  for (int d = 0; d < 16; ++d) orow[d] = o[d] * invl;
}

// ------------------------------ final 64->1 --------------------------------
__global__ void k_dot64(const float* __restrict__ X, const float* __restrict__ w,
                        float* __restrict__ y, int N) {
  int n = blockIdx.x * blockDim.x + threadIdx.x;
  if (n >= N) return;
  const float* xr = X + (size_t)n * 64;
  float acc = 0.f;
#pragma unroll
  for (int c = 0; c < 64; ++c) acc += xr[c] * w[c];
  y[n] = acc;
}

// ---------------------------------------------------------------------------
extern "C" void kernel_launch(void* const* d_in, const int* in_sizes, int n_in,
                              void* d_out, int out_size, void* d_ws, size_t ws_size,
                              hipStream_t stream) {
  (void)n_in; (void)out_size; (void)ws_size;
  const int NC = in_sizes[0] / 4;
  const int NE = in_sizes[1];
  const int NV = in_sizes[2] / 6;
  const int NT = in_sizes[3] / 768;  // == 128

  // JAX pytree flatten order (dict keys sorted alphabetically):
  // 0..3: constraint_features, edge_features, variable_features, text_hidden
  // 4:attn_out.W 5:attn_out.b | 6..19:c2v | 20..33:c2v2 | 34:cons1.W 35:cons1.b
  // 36:cons2.W 37:cons2.b | 38:cons_ln_b 39:cons_ln_g | 40:edge_ln_b 41:edge_ln_g
  // 42:k.W 43:k.b | 44:out1.W 45:out1.b | 46:out2.W | 47:q.W 48:q.b
  // 49:sem.W 50:sem.b | 51:v.W 52:v.b | 53..66:v2c | 67..80:v2c2
  // 81:var1.W 82:var1.b | 83:var2.W 84:var2.b | 85:var_ln_b 86:var_ln_g
  // 87: edge_indices (int32, [2,NE])
  // conv leaf order: edge.W final.W final.b left.W left.b ln1_b ln1_g ln2_b
  //                  ln2_g out1.W out1.b out2.W out2.b right.W
#define PF(i) ((const float*)d_in[(i)])
  const float* consF = PF(0);
  const float* varF  = PF(2);
  const float* textH = PF(3);
  const int* ei   = (const int*)d_in[87];
  const int* srcI = ei;        // constraint index per edge
  const int* dstI = ei + NE;   // variable index per edge

  // ---- workspace layout (floats) ----
  float* ws = (float*)d_ws;
  size_t off = 0;
  auto take = [&](size_t n) { float* p = ws + off; off += (n + 63) & ~(size_t)63; return p; };
  float* c0   = take((size_t)NC * 64);
  float* c1   = take((size_t)NC * 64);
  float* v0   = take((size_t)NV * 64);
  float* v1   = take((size_t)NV * 64);
  float* AB   = take((size_t)NV * 128);  // A | B ; also aliased as cat[N,128]
  float* Abuf = AB;
  float* Bbuf = AB + (size_t)NV * 64;
  float* G    = take((size_t)NV * 64);   // aggpre / hidden reuse
  float* H    = take((size_t)NV * 64);
  float* D    = take((size_t)NV);        // degrees
  float* Temb = take((size_t)128 * 64);
  float* Kt   = take((size_t)128 * 64);
  float* Vt   = take((size_t)128 * 64);
  float* evec = take(64);

  struct ConvW {
    const float *edgeW, *finalW, *finalb, *leftW, *leftb, *ln1b, *ln1g,
                *ln2b, *ln2g, *out1W, *out1b, *out2W, *out2b, *rightW;
  };
  auto loadConv = [&](int base) {
    ConvW w;
    w.edgeW = PF(base + 0);  w.finalW = PF(base + 1); w.finalb = PF(base + 2);
    w.leftW = PF(base + 3);  w.leftb  = PF(base + 4); w.ln1b   = PF(base + 5);
    w.ln1g  = PF(base + 6);  w.ln2b   = PF(base + 7); w.ln2g   = PF(base + 8);
    w.out1W = PF(base + 9);  w.out1b  = PF(base + 10);
    w.out2W = PF(base + 11); w.out2b  = PF(base + 12); w.rightW = PF(base + 13);
    return w;
  };

  auto gemm = [&](const float* X, int N, int K, const float* W, const float* b,
                  const float* deg, const float* b2, float* Y, int relu) {
    int blocks = (N + 127) / 128;
    k_gemm64<<<blocks, 256, 0, stream>>>(X, N, K, W, b, deg, b2, Y, relu);
  };

  const float* elnb = PF(40);
  auto runConv = [&](const ConvW& w, const float* tgtCur, float* tgtNew, int Ntgt,
                     const float* srcCur, int Nsrc, int tgt_is_src) {
    k_evec<<<1, 64, 0, stream>>>(elnb, w.edgeW, evec);
    gemm(tgtCur, Ntgt, 64, w.leftW, w.leftb, nullptr, nullptr, Abuf, 0);   // L
    gemm(srcCur, Nsrc, 64, w.rightW, nullptr, nullptr, nullptr, Bbuf, 0);  // R
    long long nAgg = (long long)Ntgt * 64;
    k_fill<<<(int)((nAgg + 255) / 256), 256, 0, stream>>>(G, nAgg, 0.f);
    k_fill<<<(Ntgt + 255) / 256, 256, 0, stream>>>(D, (long long)Ntgt, 0.f);
    k_edge<<<(int)(((long long)NE * 8 + 255) / 256), 256, 0, stream>>>(
        srcI, dstI, NE, Abuf, Bbuf, evec, w.ln1g, w.ln1b, G, D, tgt_is_src);
    gemm(G, Ntgt, 64, w.finalW, nullptr, D, w.finalb, H, 0);               // F
    k_lncat<<<(int)(((long long)Ntgt * 8 + 255) / 256), 256, 0, stream>>>(
        H, tgtCur, w.ln2g, w.ln2b, AB, Ntgt);                              // cat
    gemm(AB, Ntgt, 128, w.out1W, w.out1b, nullptr, nullptr, G, 1);         // hidden
    gemm(G, Ntgt, 64, w.out2W, w.out2b, nullptr, nullptr, tgtNew, 0);      // new tgt
  };

  // ---- input embeddings ----
  k_embed1<<<(int)(((long long)NC * 64 + 255) / 256), 256, 0, stream>>>(
      consF, 4, NC, PF(39), PF(38), PF(34), PF(35), Abuf);
  gemm(Abuf, NC, 64, PF(36), PF(37), nullptr, nullptr, c0, 1);
  k_embed1<<<(int)(((long long)NV * 64 + 255) / 256), 256, 0, stream>>>(
      varF, 6, NV, PF(86), PF(85), PF(81), PF(82), Abuf);
  gemm(Abuf, NV, 64, PF(83), PF(84), nullptr, nullptr, v0, 1);

  // ---- four graph convolutions ----
  ConvW pv2c = loadConv(53), pc2v = loadConv(6), pv2c2 = loadConv(67), pc2v2 = loadConv(20);
  runConv(pv2c,  c0, c1, NC, v0, NV, 1);  // c = conv(v -> c)
  runConv(pc2v,  v0, v1, NV, c1, NC, 0);  // v = conv(c -> v)
  runConv(pv2c2, c1, c0, NC, v1, NV, 1);  // c = conv(v -> c)
  runConv(pc2v2, v1, v0, NV, c0, NC, 0);  // v = conv(c -> v)

  // ---- cross attention over text tokens ----
  gemm(textH, NT, 768, PF(49), PF(50), nullptr, nullptr, Temb, 0);  // sem
  gemm(Temb, NT, 64, PF(42), PF(43), nullptr, nullptr, Kt, 0);      // K
  gemm(Temb, NT, 64, PF(51), PF(52), nullptr, nullptr, Vt, 0);      // V
  gemm(v0,   NV, 64, PF(47), PF(48), nullptr, nullptr, Abuf, 0);    // Q
  k_attn<<<(int)(((long long)NV * 4 + 255) / 256), 256, 0, stream>>>(
      Abuf, Kt, Vt, Bbuf, NV);
  gemm(Bbuf, NV, 64, PF(4),  PF(5),  nullptr, nullptr, H, 0);       // attn_out
  gemm(H,    NV, 64, PF(44), PF(45), nullptr, nullptr, Abuf, 1);    // out1 + ReLU
  k_dot64<<<(NV + 255) / 256, 256, 0, stream>>>(Abuf, PF(46), (float*)d_out, NV);
#undef PF
}